// RotateRelEbd_45724221833316
// MI455X (gfx1250) — compile-verified
//
#include <hip/hip_runtime.h>

// out[b,c,h,w] = x[b,c,h,w] + circle_mats[min(h, w, H-1-h, W-1-w)][c]
// HBM-streaming elementwise op: 805 MB total traffic -> ~34.6 us floor at
// 23.3 TB/s. One block per (b,c) slab (64KB in + 64KB out) to amortize the
// per-wave SALU prologue; h (and thus rowmin) is a compile-time constant in
// each unrolled iteration, so edge-row selects mostly fold away.

typedef float v4f __attribute__((ext_vector_type(4)));

constexpr int B = 8, C = 768, H = 8, W = 2048;
constexpr int TPB   = 256;                 // 8 wave32 waves per block
constexpr int SLAB  = H * W;               // 16384 floats per (b,c) slab
constexpr int ITERS = SLAB / 4 / TPB;      // 16 float4-iterations per thread

__global__ __launch_bounds__(TPB) void circle_edge_add_kernel(
    const float* __restrict__ x,
    const float* __restrict__ circle,   // (4, C)
    float* __restrict__ out)
{
    const int tid = threadIdx.x;
    const int c   = blockIdx.x;         // 0..767, block-uniform
    const int b   = blockIdx.y;         // 0..7

    // Block-uniform table gathers -> scalar s_load clause.
    const float v0 = circle[0 * C + c];
    const float v1 = circle[1 * C + c];
    const float v2 = circle[2 * C + c];
    const float v3 = circle[3 * C + c];

    const size_t base = ((size_t)b * C + c) * (size_t)SLAB;

    // gfx1250 prefetch (global_prefetch_b8): pull the slab's second half.
    // Always in-bounds (within this block's own 64KB slab).
    __builtin_prefetch(x + base + (size_t)(SLAB / 2) + tid * 4, 0, 0);

#pragma unroll
    for (int it = 0; it < ITERS; ++it) {
        const int idx = it * TPB + tid;       // float4 index within slab, 0..4095
        const int h   = it >> 1;              // compile-time constant per iteration
        const int rowmin = (h < H - 1 - h) ? h : (H - 1 - h);   // constant 0..3
        const int w4  = idx & (W / 4 - 1);    // 0..511
        const size_t off = base + (size_t)idx * 4;

        v4f xv = __builtin_nontemporal_load((const v4f*)(x + off));

        const int j0 = w4 * 4;
#pragma unroll
        for (int k = 0; k < 4; ++k) {
            const int j = j0 + k;
            const int d = min(rowmin, min(j, (W - 1) - j));     // 0..3
            const float a = (d == 0) ? v0 : (d == 1) ? v1 : (d == 2) ? v2 : v3;
            xv[k] += a;
        }

        __builtin_nontemporal_store(xv, (v4f*)(out + off));
    }
}

extern "C" void kernel_launch(void* const* d_in, const int* in_sizes, int n_in,
                              void* d_out, int out_size, void* d_ws, size_t ws_size,
                              hipStream_t stream) {
    const float* x      = (const float*)d_in[0];   // (B, C, H, W) fp32
    const float* circle = (const float*)d_in[1];   // (4, C) fp32
    float* out          = (float*)d_out;           // (B, C, H, W) fp32

    dim3 grid(C, B);                               // (768, 8) = 6144 blocks
    circle_edge_add_kernel<<<grid, TPB, 0, stream>>>(x, circle, out);
}